// QuestionAttention_22325240004679
// MI455X (gfx1250) — compile-verified
//
#include <hip/hip_runtime.h>
#include <hip/hip_bf16.h>
#include <math.h>

// Problem constants (match reference)
#define B_   16
#define S_   2048
#define D_   2048
#define BSZ  (B_ * S_)          // 32768 rows of [B*S, D]

#define WAVES        8
#define TPB          (WAVES * 32)        // 256 threads = 8 wave32
#define NT_PER_WAVE  (D_ / (WAVES * 16)) // 16 N-tiles of 16 cols per wave
#define NPAIRS       (NT_PER_WAVE / 2)   // processed two tiles per K-pass
#define SG_PITCH     2052                // floats; +4 pad -> conflict-free ds_load_b64

typedef float v2f __attribute__((ext_vector_type(2)));
typedef float v8f __attribute__((ext_vector_type(8)));

// Branchless fast tanh: 1 - 2/(exp(2x)+1); exact saturation via exp over/underflow.
__device__ __forceinline__ float fast_tanh(float x) {
  float t = __expf(2.0f * x);
  return 1.0f - 2.0f / (t + 1.0f);
}

// ---------------------------------------------------------------------------
// Kernel 0: mq = tanh(Wm @ m_q + bm)   (tiny: 2048 dot products of length 2048)
// ---------------------------------------------------------------------------
__global__ __launch_bounds__(TPB)
void mq_kernel(const float* __restrict__ Wm, const float* __restrict__ m_q,
               const float* __restrict__ bm, float* __restrict__ mq_out) {
  int e = blockIdx.x * blockDim.x + threadIdx.x;
  if (e >= D_) return;
  const float* row = Wm + (size_t)e * D_;
  float acc = bm[e];
  for (int d = 0; d < D_; d += 4) {
    acc += row[d]     * m_q[d];
    acc += row[d + 1] * m_q[d + 1];
    acc += row[d + 2] * m_q[d + 2];
    acc += row[d + 3] * m_q[d + 3];
  }
  mq_out[e] = fast_tanh(acc);
}

// ---------------------------------------------------------------------------
// Kernel 1: fully fused per 16-row tile of [B*S, D]:
//   Sg = tanh(H Wq^T + bq) * mq        (GEMM1, f32 WMMA, Sg -> LDS)
//   logits = Sg Wa^T + ba              (GEMM2, f32 WMMA, A from LDS)
//   online softmax over D fused with sum(alpha * hidden)
//   out[r] = o/s ; m_q_new[r] = m_q[r % S] + out[r]
// ---------------------------------------------------------------------------
__global__ __launch_bounds__(TPB)
void fused_attn_kernel(const float* __restrict__ hidden,
                       const float* __restrict__ m_q,
                       const float* __restrict__ Wq, const float* __restrict__ bq,
                       const float* __restrict__ Wa, const float* __restrict__ ba,
                       const float* __restrict__ mq,
                       float* __restrict__ out) {
  __shared__ float sg[16 * SG_PITCH];      // ~131 KB: Sg tile [16 x 2048] (padded)
  __shared__ float red_m[WAVES][16];
  __shared__ float red_s[WAVES][16];
  __shared__ float red_o[WAVES][16];

  const int lane = threadIdx.x & 31;
  const int wave = threadIdx.x >> 5;
  const int half = lane >> 4;     // A/B fragment: lanes 0-15 hold K={0,1}, 16-31 K={2,3}
  const int l16  = lane & 15;
  const int ksub = 2 * half;
  const int nbase = wave * (NT_PER_WAVE * 16);   // wave's 256-column range

  const size_t r0   = (size_t)blockIdx.x * 16;             // first flattened row
  const float* Arow = hidden + (r0 + (size_t)l16) * D_;    // this lane's A row

  // ---------------- Phase 1: GEMM1 + tanh + gate -> LDS ----------------
  for (int p = 0; p < NPAIRS; ++p) {
    const int n0 = nbase + p * 32;                         // two adjacent 16-col tiles
    const float* Brow0 = Wq + (size_t)(n0 + l16)      * D_;
    const float* Brow1 = Wq + (size_t)(n0 + 16 + l16) * D_;
    if (p + 1 < NPAIRS) {                                  // warm cache for next pair
      __builtin_prefetch(Wq + (size_t)(n0 + 32 + l16) * D_, 0, 0);
      __builtin_prefetch(Wq + (size_t)(n0 + 48 + l16) * D_, 0, 0);
    }
    v8f c0 = {}, c1 = {};
    for (int k = 0; k < D_; k += 4) {                      // 1 A-load feeds 2 WMMAs
      v2f a  = *(const v2f*)(Arow  + k + ksub);
      v2f b0 = *(const v2f*)(Brow0 + k + ksub);
      v2f b1 = *(const v2f*)(Brow1 + k + ksub);
      c0 = __builtin_amdgcn_wmma_f32_16x16x4_f32(false, a, false, b0, (short)0, c0,
                                                 false, false);
      c1 = __builtin_amdgcn_wmma_f32_16x16x4_f32(false, a, false, b1, (short)0, c1,
                                                 false, false);
    }
    const int   nc0 = n0 + l16,      nc1 = n0 + 16 + l16;
    const float bi0 = bq[nc0],       bi1 = bq[nc1];
    const float g0  = mq[nc0],       g1  = mq[nc1];
#pragma unroll
    for (int i = 0; i < 8; ++i) {
      const int row = i + 8 * half;                 // C layout: VGPR i, halves M / M+8
      sg[row * SG_PITCH + nc0] = fast_tanh(c0[i] + bi0) * g0;
      sg[row * SG_PITCH + nc1] = fast_tanh(c1[i] + bi1) * g1;
    }
  }
  __syncthreads();

  // -------- Phase 2: GEMM2 + online softmax + weighted hidden sum --------
  float m[8], s[8], o[8];
#pragma unroll
  for (int i = 0; i < 8; ++i) { m[i] = -INFINITY; s[i] = 0.f; o[i] = 0.f; }

  const float* Asg = &sg[l16 * SG_PITCH + ksub];           // LDS A fragment base

  for (int p = 0; p < NPAIRS; ++p) {
    const int n0 = nbase + p * 32;
    const float* Brow0 = Wa + (size_t)(n0 + l16)      * D_;
    const float* Brow1 = Wa + (size_t)(n0 + 16 + l16) * D_;
    if (p + 1 < NPAIRS) {
      __builtin_prefetch(Wa + (size_t)(n0 + 32 + l16) * D_, 0, 0);
      __builtin_prefetch(Wa + (size_t)(n0 + 48 + l16) * D_, 0, 0);
    }
    v8f c0 = {}, c1 = {};
    for (int k = 0; k < D_; k += 4) {
      v2f a  = *(const v2f*)(Asg + k);                     // ds_load_b64, padded pitch
      v2f b0 = *(const v2f*)(Brow0 + k + ksub);
      v2f b1 = *(const v2f*)(Brow1 + k + ksub);
      c0 = __builtin_amdgcn_wmma_f32_16x16x4_f32(false, a, false, b0, (short)0, c0,
                                                 false, false);
      c1 = __builtin_amdgcn_wmma_f32_16x16x4_f32(false, a, false, b1, (short)0, c1,
                                                 false, false);
    }
    const int   nc0 = n0 + l16,  nc1 = n0 + 16 + l16;
    const float ba0 = ba[nc0],   ba1 = ba[nc1];
#pragma unroll
    for (int i = 0; i < 8; ++i) {
      const int row = i + 8 * half;
      // tile 0 element
      {
        const float e  = c0[i] + ba0;
        const float h  = hidden[(r0 + row) * (size_t)D_ + nc0];  // L2/L0 hit
        const float mn = fmaxf(m[i], e);
        const float r  = __expf(m[i] - mn);
        const float w  = __expf(e - mn);
        s[i] = s[i] * r + w;
        o[i] = o[i] * r + w * h;
        m[i] = mn;
      }
      // tile 1 element
      {
        const float e  = c1[i] + ba1;
        const float h  = hidden[(r0 + row) * (size_t)D_ + nc1];
        const float mn = fmaxf(m[i], e);
        const float r  = __expf(m[i] - mn);
        const float w  = __expf(e - mn);
        s[i] = s[i] * r + w;
        o[i] = o[i] * r + w * h;
        m[i] = mn;
      }
    }
  }

  // Cross-lane online-softmax merge within each 16-lane half (cols of the tile)
#pragma unroll
  for (int mask = 1; mask < 16; mask <<= 1) {
#pragma unroll
    for (int i = 0; i < 8; ++i) {
      const float m2 = __shfl_xor(m[i], mask, 32);
      const float s2 = __shfl_xor(s[i], mask, 32);
      const float o2 = __shfl_xor(o[i], mask, 32);
      const float mn = fmaxf(m[i], m2);
      const float r1 = __expf(m[i] - mn);
      const float r2 = __expf(m2 - mn);
      s[i] = s[i] * r1 + s2 * r2;
      o[i] = o[i] * r1 + o2 * r2;
      m[i] = mn;
    }
  }

  if (l16 == 0) {
#pragma unroll
    for (int i = 0; i < 8; ++i) {
      const int row = i + 8 * half;
      red_m[wave][row] = m[i];
      red_s[wave][row] = s[i];
      red_o[wave][row] = o[i];
    }
  }
  __syncthreads();

  // Cross-wave merge + final outputs (16 rows)
  if (threadIdx.x < 16) {
    const int row = threadIdx.x;
    float mt = -INFINITY, st = 0.f, ot = 0.f;
    for (int w = 0; w < WAVES; ++w) {
      const float m2 = red_m[w][row];
      const float s2 = red_s[w][row];
      const float o2 = red_o[w][row];
      const float mn = fmaxf(mt, m2);
      const float r1 = __expf(mt - mn);
      const float r2 = __expf(m2 - mn);
      st = st * r1 + s2 * r2;
      ot = ot * r1 + o2 * r2;
      mt = mn;
    }
    const float res = ot / st;
    const size_t r  = r0 + row;
    out[r]       = res;                        // output[b,s]
    out[BSZ + r] = m_q[r % S_] + res;          // m_q_new[b,s] = m_q[s] + output
  }
}

// ---------------------------------------------------------------------------
extern "C" void kernel_launch(void* const* d_in, const int* in_sizes, int n_in,
                              void* d_out, int out_size, void* d_ws, size_t ws_size,
                              hipStream_t stream) {
  const float* hidden = (const float*)d_in[0];
  const float* m_q    = (const float*)d_in[1];
  const float* Wq     = (const float*)d_in[2];
  const float* bq     = (const float*)d_in[3];
  const float* Wm     = (const float*)d_in[4];
  const float* bm     = (const float*)d_in[5];
  const float* Wa     = (const float*)d_in[6];
  const float* ba     = (const float*)d_in[7];
  float* out = (float*)d_out;
  float* mq  = (float*)d_ws;                   // D_ floats of scratch

  mq_kernel<<<D_ / TPB, TPB, 0, stream>>>(Wm, m_q, bm, mq);
  fused_attn_kernel<<<BSZ / 16, TPB, 0, stream>>>(hidden, m_q, Wq, bq, Wa, ba, mq, out);
}